// MultiModalitiesAttnMashupTpmHead_17532056502514
// MI455X (gfx1250) — compile-verified
//
#include <hip/hip_runtime.h>
#include <hip/hip_bf16.h>
#include <math.h>

// ---------------------------------------------------------------------------
// MI455X (gfx1250) implementation. wave32, WMMA 16x16x4 F32 for all GEMMs
// (fp32 reference -> keep fp32 MACs on the matrix pipe: 2048 FLOP/instr).
// GEMM: 8-wave block computes 128x64; B tile staged in LDS with k-pair
// interleaving (ds_load_b64 fragments, conflict-free banking); branchless
// 8x4 WMMA hot loop per 32-deep K block; global_prefetch of next K block.
// ---------------------------------------------------------------------------

#define BATCH 1024
#define EDIM  256
#define HEADS 8
#define NMEM  128
#define NOTH  6
#define NMOD  8

typedef __attribute__((ext_vector_type(2))) float v2f;
typedef __attribute__((ext_vector_type(8))) float v8f;

// ---------------------------------------------------------------------------
// C[M,N] = act(A[M,K] @ B + bias)
//   TRANSB == 0 : B is [K,N] row-major (y = x @ w)
//   TRANSB == 1 : B is [N,K] row-major (y = x @ w^T, torch Linear / MHA in_w)
// Requires M%128==0, N%64==0, K%4==0 (true for all shapes used here).
//
// LDS B tile layout: pair p = k/2 (16 pair-rows), 64 cols; Bs[p*160 + 2n + par]
// holds B element (k = 2p + par, n). One ds_load_b64 at (p*160 + 2n) yields the
// (k, k+1) fragment pair directly into an even-aligned VGPR pair. Pair-row
// stride 160 dwords == 32 mod 64 banks -> the two WMMA lane halves (p vs p+1)
// read opposite 32-bank halves: conflict-free.
// ---------------------------------------------------------------------------
template <int TRANSB, int RELU>
__global__ __launch_bounds__(256) void gemm_wmma_f32(
    const float* __restrict__ A, const float* __restrict__ Bm,
    const float* __restrict__ bias, float* __restrict__ C,
    int N, int K, int tilesN)
{
  __shared__ float Bs[16 * 160];   // 10 KB

  int tid  = threadIdx.x;
  int wave = tid >> 5;
  int lane = tid & 31;
  int bm = blockIdx.x / tilesN;
  int bn = blockIdx.x - bm * tilesN;
  int row0 = bm * 128 + wave * 16;
  int col0 = bn << 6;
  int hl = lane >> 4;        // lane half -> K pair (k4..k4+1 vs k4+2..k4+3)
  int mr = lane & 15;        // M row (A) / N col (B,C)

  v8f acc[4];
#pragma unroll
  for (int t = 0; t < 4; ++t)
#pragma unroll
    for (int i = 0; i < 8; ++i) acc[t][i] = 0.f;

  const float* Arow = A + (size_t)(row0 + mr) * K + 2 * hl;

  for (int kb = 0; kb < K; kb += 32) {
    int KB = (K - kb < 32) ? (K - kb) : 32;   // multiple of 4

    __syncthreads();
    if (TRANSB) {
      // B[n][k]: 4 threads per n, each copies 8 consecutive k = 4 k-pairs.
      int n = tid >> 2;
      int kbase = (tid & 3) << 3;
      const float* src = Bm + (size_t)(col0 + n) * K + kb + kbase;
#pragma unroll
      for (int j = 0; j < 4; ++j) {
        if (kbase + 2 * j < KB) {
          v2f pv;
          pv.x = src[2 * j];
          pv.y = src[2 * j + 1];
          *(v2f*)&Bs[((kbase >> 1) + j) * 160 + 2 * n] = pv;
        }
      }
      if (kb + 32 < K) __builtin_prefetch(src + 32, 0, 0);
    } else {
      // B[k][n]: thread covers pair id q = tid + 256*j (p = q/64, n = q%64);
      // both global loads are 64-wide coalesced row segments.
#pragma unroll
      for (int j = 0; j < 4; ++j) {
        int q = tid + 256 * j;
        int p = q >> 6;
        int n = q & 63;
        if (2 * p < KB) {
          v2f pv;
          pv.x = Bm[(size_t)(kb + 2 * p) * N + col0 + n];
          pv.y = Bm[(size_t)(kb + 2 * p + 1) * N + col0 + n];
          *(v2f*)&Bs[p * 160 + 2 * n] = pv;
        }
      }
      if (kb + 32 < K)
        __builtin_prefetch(Bm + (size_t)(kb + 32 + (tid >> 6)) * N + col0 + (tid & 63), 0, 0);
    }
    __syncthreads();

    if (KB == 32) {
      // Hot path: branchless. Preload the K-block's 8 A fragments (b64 each).
      v2f af[8];
#pragma unroll
      for (int s = 0; s < 8; ++s) {
        af[s].x = Arow[kb + 4 * s];
        af[s].y = Arow[kb + 4 * s + 1];
      }
      if (kb + 32 < K) __builtin_prefetch(Arow + kb + 32, 0, 0);
#pragma unroll
      for (int s = 0; s < 8; ++s) {
        int pbase = 2 * s + hl;              // pair index of (k4 + 2*hl)
#pragma unroll
        for (int t = 0; t < 4; ++t) {
          v2f b = *(const v2f*)&Bs[pbase * 160 + 2 * ((t << 4) + mr)];
          acc[t] = __builtin_amdgcn_wmma_f32_16x16x4_f32(
              false, af[s], false, b, (short)0, acc[t], false, false);
        }
      }
    } else {
      // Tail path (only K=300: KB=12 -> 3 steps).
      for (int s = 0; 4 * s < KB; ++s) {
        v2f a;
        a.x = Arow[kb + 4 * s];
        a.y = Arow[kb + 4 * s + 1];
        int pbase = 2 * s + hl;
#pragma unroll
        for (int t = 0; t < 4; ++t) {
          v2f b = *(const v2f*)&Bs[pbase * 160 + 2 * ((t << 4) + mr)];
          acc[t] = __builtin_amdgcn_wmma_f32_16x16x4_f32(
              false, a, false, b, (short)0, acc[t], false, false);
        }
      }
    }
  }

#pragma unroll
  for (int t = 0; t < 4; ++t) {
    int n = col0 + (t << 4) + mr;
    float bs = bias ? bias[n] : 0.f;
#pragma unroll
    for (int i = 0; i < 8; ++i) {
      int r = row0 + i + 8 * hl;              // C layout: M = vgpr + 8*laneHalf
      float v = acc[t][i] + bs;
      if (RELU) v = fmaxf(v, 0.f);
      C[(size_t)r * N + n] = v;
    }
  }
}

static void launch_gemm(hipStream_t s, const float* A, const float* B,
                        const float* bias, float* C, int M, int N, int K,
                        int transB, int relu)
{
  int tilesN = N >> 6;
  int blocks = (M >> 7) * tilesN;             // exact grid: no early-outs
  if (transB) {
    if (relu) gemm_wmma_f32<1, 1><<<blocks, 256, 0, s>>>(A, B, bias, C, N, K, tilesN);
    else      gemm_wmma_f32<1, 0><<<blocks, 256, 0, s>>>(A, B, bias, C, N, K, tilesN);
  } else {
    if (relu) gemm_wmma_f32<0, 1><<<blocks, 256, 0, s>>>(A, B, bias, C, N, K, tilesN);
    else      gemm_wmma_f32<0, 0><<<blocks, 256, 0, s>>>(A, B, bias, C, N, K, tilesN);
  }
}

// ---------------------------------------------------------------------------
// Protein neighbor gather + per-hop mean/max -> agg_in [B, 4E]
// block = b, 256 threads = embedding column. Table (19.5MB) is L2-resident.
// ---------------------------------------------------------------------------
__global__ __launch_bounds__(256) void gather_agg_kernel(
    const float* __restrict__ emb, const int* __restrict__ neighbors,
    float* __restrict__ agg_in)
{
  int b = blockIdx.x, c = threadIdx.x;
#pragma unroll
  for (int hop = 0; hop < 2; ++hop) {
    const int* nb = neighbors + ((size_t)hop * BATCH + b) * NMEM;
    float s = 0.f, mx = -INFINITY;
    for (int j = 0; j < NMEM; ++j) {
      float v = emb[(size_t)nb[j] * EDIM + c];
      s += v;
      mx = fmaxf(mx, v);
    }
    float* o = agg_in + (size_t)b * (4 * EDIM) + hop * (2 * EDIM);
    o[c] = s * (1.f / NMEM);
    o[EDIM + c] = mx;
  }
}

// ---------------------------------------------------------------------------
// LayerNorm over 256 columns, block per row.
// ---------------------------------------------------------------------------
__global__ __launch_bounds__(256) void layernorm256_kernel(
    const float* __restrict__ x, const float* __restrict__ g,
    const float* __restrict__ b, float* __restrict__ y)
{
  __shared__ float red[256];
  int row = blockIdx.x, t = threadIdx.x;
  float v = x[(size_t)row * 256 + t];
  red[t] = v;
  __syncthreads();
  for (int s = 128; s > 0; s >>= 1) {
    if (t < s) red[t] += red[t + s];
    __syncthreads();
  }
  float mean = red[0] * (1.f / 256.f);
  __syncthreads();
  float d = v - mean;
  red[t] = d * d;
  __syncthreads();
  for (int s = 128; s > 0; s >>= 1) {
    if (t < s) red[t] += red[t + s];
    __syncthreads();
  }
  float var = red[0] * (1.f / 256.f);
  y[(size_t)row * 256 + t] = d * rsqrtf(var + 1e-5f) * g[t] + b[t];
}

// ---------------------------------------------------------------------------
// Row L2 normalize (block per row, generic K, safe in-place).
// ---------------------------------------------------------------------------
__global__ __launch_bounds__(256) void rownorm_kernel(
    const float* __restrict__ x, float* __restrict__ y, int K, float clampMin)
{
  __shared__ float red[256];
  int row = blockIdx.x, t = threadIdx.x;
  const float* xr = x + (size_t)row * K;
  float ss = 0.f;
  for (int i = t; i < K; i += 256) { float v = xr[i]; ss += v * v; }
  red[t] = ss;
  __syncthreads();
  for (int s = 128; s > 0; s >>= 1) {
    if (t < s) red[t] += red[t + s];
    __syncthreads();
  }
  float inv = 1.f / fmaxf(sqrtf(red[0]), clampMin);
  float* yr = y + (size_t)row * K;
  for (int i = t; i < K; i += 256) yr[i] = xr[i] * inv;
}

// ---------------------------------------------------------------------------
// Tiny MHA core: one wave per (b,h); lane = head dim (d = 32 = wave32).
// qkv: [L, B, 3E] with q|k|v at col offsets 0|256|512, head h at h*32.
// o:   [L, B, E]. kpm: [B, L] ints, nonzero = masked (score := -1e9).
// ---------------------------------------------------------------------------
template <int L>
__global__ __launch_bounds__(256) void attn_small_kernel(
    const float* __restrict__ qkv, float* __restrict__ o,
    const int* __restrict__ kpm, int Bn)
{
  int gw = (int)((blockIdx.x * blockDim.x + threadIdx.x) >> 5);
  int lane = threadIdx.x & 31;
  if (gw >= Bn * HEADS) return;
  int b = gw >> 3, h = gw & 7;
  const float scale = 0.17677669529663687f;  // 1/sqrt(32)

  float q[L], k[L], v[L];
  bool msk[L];
#pragma unroll
  for (int l = 0; l < L; ++l) {
    const float* base = qkv + ((size_t)l * Bn + b) * (3 * EDIM) + h * 32 + lane;
    q[l] = base[0] * scale;
    k[l] = base[EDIM];
    v[l] = base[2 * EDIM];
    msk[l] = kpm ? (kpm[b * L + l] != 0) : false;
  }
#pragma unroll
  for (int ql = 0; ql < L; ++ql) {
    float s[L];
    float mx = -INFINITY;
#pragma unroll
    for (int kl = 0; kl < L; ++kl) {
      float p = q[ql] * k[kl];
      for (int off = 16; off > 0; off >>= 1) p += __shfl_xor(p, off, 32);
      s[kl] = msk[kl] ? -1e9f : p;
      mx = fmaxf(mx, s[kl]);
    }
    float denom = 0.f, ov = 0.f;
#pragma unroll
    for (int kl = 0; kl < L; ++kl) {
      float e = expf(s[kl] - mx);
      denom += e;
      ov += e * v[kl];
    }
    o[((size_t)ql * Bn + b) * EDIM + h * 32 + lane] = ov / denom;
  }
}

// cell_emb = mean over L=2 of attn output -> write into xc[:, col0:col0+256]
__global__ __launch_bounds__(256) void mean2_kernel(
    const float* __restrict__ x, float* __restrict__ y, int ldy, int col0, int Bn)
{
  int b = blockIdx.x, e = threadIdx.x;
  y[(size_t)b * ldy + col0 + e] =
      0.5f * (x[(size_t)b * EDIM + e] + x[((size_t)Bn + b) * EDIM + e]);
}

// masked mean pool over L=8 modality axis -> xc[:, col0:col0+256]
__global__ __launch_bounds__(256) void masked_pool_kernel(
    const float* __restrict__ x, const int* __restrict__ mask,
    float* __restrict__ y, int ldy, int col0, int Bn)
{
  int b = blockIdx.x, e = threadIdx.x;
  float s = 0.f, w = 0.f;
#pragma unroll
  for (int l = 0; l < NMOD; ++l) {
    float wl = mask[b * NMOD + l] ? 0.f : 1.f;
    s += x[((size_t)l * Bn + b) * EDIM + e] * wl;
    w += wl;
  }
  y[(size_t)b * ldy + col0 + e] = s / w;   // >=2 slots always unmasked
}

// random single-modality contrastive view: xcv[b] = [heads[ra], tails[rb], cell[ci]]
__global__ __launch_bounds__(256) void build_view_kernel(
    const float* __restrict__ heads_all, const float* __restrict__ tails_all,
    const float* __restrict__ cell_all, const int* __restrict__ ra,
    const int* __restrict__ rb, const int* __restrict__ ci,
    float* __restrict__ xcv, int Bn)
{
  int b = blockIdx.x, e = threadIdx.x;
  float* o = xcv + (size_t)b * (3 * EDIM);
  o[e]            = heads_all[((size_t)ra[b] * Bn + b) * EDIM + e];
  o[EDIM + e]     = tails_all[((size_t)rb[b] * Bn + b) * EDIM + e];
  o[2 * EDIM + e] = cell_all [((size_t)ci[b] * Bn + b) * EDIM + e];
}

// N=1 classifier head: one wave per row. y = scale*((x.w + bias) + add)
__global__ __launch_bounds__(256) void gemv_rows_kernel(
    const float* __restrict__ X, const float* __restrict__ w,
    const float* __restrict__ bias, const float* __restrict__ add,
    float scale, float* __restrict__ y, int K, int rows)
{
  int gw = (int)((blockIdx.x * blockDim.x + threadIdx.x) >> 5);
  int lane = threadIdx.x & 31;
  if (gw >= rows) return;
  const float* xr = X + (size_t)gw * K;
  float s = 0.f;
  for (int i = lane; i < K; i += 32) s += xr[i] * w[i];
  for (int off = 16; off > 0; off >>= 1) s += __shfl_xor(s, off, 32);
  if (lane == 0) {
    float v = s + bias[0];
    if (add) v += add[gw];
    y[gw] = v * scale;
  }
}

// per-row: loss = logsumexp(row) - row[row]  (max-shifted)
__global__ __launch_bounds__(256) void row_lse_loss_kernel(
    const float* __restrict__ logits, float* __restrict__ loss, int N)
{
  __shared__ float red[256];
  int row = blockIdx.x, t = threadIdx.x;
  const float* xr = logits + (size_t)row * N;
  float mx = -INFINITY;
  for (int i = t; i < N; i += 256) mx = fmaxf(mx, xr[i]);
  red[t] = mx;
  __syncthreads();
  for (int s = 128; s > 0; s >>= 1) {
    if (t < s) red[t] = fmaxf(red[t], red[t + s]);
    __syncthreads();
  }
  mx = red[0];
  __syncthreads();
  float se = 0.f;
  for (int i = t; i < N; i += 256) se += expf(xr[i] - mx);
  red[t] = se;
  __syncthreads();
  for (int s = 128; s > 0; s >>= 1) {
    if (t < s) red[t] += red[t + s];
    __syncthreads();
  }
  if (t == 0) loss[row] = logf(red[0]) + mx - xr[row];
}

__global__ __launch_bounds__(256) void mean_reduce_kernel(
    const float* __restrict__ x, float* __restrict__ out, int n)
{
  __shared__ float red[256];
  int t = threadIdx.x;
  float s = 0.f;
  for (int i = t; i < n; i += 256) s += x[i];
  red[t] = s;
  __syncthreads();
  for (int k = 128; k > 0; k >>= 1) {
    if (t < k) red[t] += red[t + k];
    __syncthreads();
  }
  if (t == 0) out[0] = red[0] / (float)n;
}

// ---------------------------------------------------------------------------
// Host orchestration. d_in order = params leaves (dict order) then inputs.
// ---------------------------------------------------------------------------
extern "C" void kernel_launch(void* const* d_in, const int* in_sizes, int n_in,
                              void* d_out, int out_size, void* d_ws, size_t ws_size,
                              hipStream_t stream)
{
  (void)in_sizes; (void)n_in; (void)out_size; (void)ws_size;

  const float* protein_emb = (const float*)d_in[0];   // [19000,256]
  const float* agg_w   = (const float*)d_in[1];       // [1024,256]
  const float* agg_b   = (const float*)d_in[2];
  const float* ln_g    = (const float*)d_in[3];
  const float* ln_b    = (const float*)d_in[4];
  const float* mashup_w= (const float*)d_in[5];       // [800,256]
  const float* mashup_b= (const float*)d_in[6];
  const float* t_w     = (const float*)d_in[7];       // [768,256]
  const float* t_b     = (const float*)d_in[8];
  const float* g_w     = (const float*)d_in[9];       // [300,256]
  const float* g_b     = (const float*)d_in[10];
  const float* modal_w = (const float*)d_in[11];      // [6,512,256]
  const float* modal_bias = (const float*)d_in[12];   // [6,256]
  const float* ca_in_w = (const float*)d_in[13];      // [768,256]
  const float* ca_in_b = (const float*)d_in[14];
  const float* ca_out_w= (const float*)d_in[15];      // [256,256]
  const float* ca_out_b= (const float*)d_in[16];
  const float* ma_in_w = (const float*)d_in[17];
  const float* ma_in_b = (const float*)d_in[18];
  const float* ma_out_w= (const float*)d_in[19];
  const float* ma_out_b= (const float*)d_in[20];
  const float* c1_w    = (const float*)d_in[21];      // [768,512]
  const float* c1_b    = (const float*)d_in[22];
  const float* c2a_w   = (const float*)d_in[23];      // [512,256]
  const float* c2a_b   = (const float*)d_in[24];
  const float* c2b_w   = (const float*)d_in[25];      // [256,1]
  const float* c2b_b   = (const float*)d_in[26];
  const float* drug_a    = (const float*)d_in[27];    // [1024,768]
  const float* dv_drug_a = (const float*)d_in[28];    // [1024,300]
  const float* drug_b    = (const float*)d_in[29];
  const float* dv_drug_b = (const float*)d_in[30];
  const float* cell_mashup = (const float*)d_in[31];  // [1024,800]
  const float* modal_a   = (const float*)d_in[32];    // [6,1024,512]
  const float* modal_b_in= (const float*)d_in[33];
  const int* neighbors  = (const int*)d_in[34];       // [2,1024,128]
  const int* heads_mask = (const int*)d_in[35];       // [1024,8]
  const int* tail_mask  = (const int*)d_in[36];
  const int* cell_idx1  = (const int*)d_in[37];
  const int* cell_idx2  = (const int*)d_in[38];
  const int* ra1 = (const int*)d_in[39];
  const int* rb1 = (const int*)d_in[40];
  const int* ra2 = (const int*)d_in[41];
  const int* rb2 = (const int*)d_in[42];

  float* out = (float*)d_out;  // [0,1024)=out, [1024]=cos, [1025,2049)=sub

  // workspace carve (floats)
  float* ws = (float*)d_ws;
  size_t off = 0;
  auto take = [&](size_t n) { float* p = ws + off; off += n; return p; };
  float* agg_in   = take((size_t)1024 * 1024);
  float* mash_n   = take((size_t)1024 * 800);
  float* pre_cell = take((size_t)1024 * 256);
  float* cell_all = take((size_t)2 * 1024 * 256);
  float* qkv_cell = take((size_t)2048 * 768);
  float* o_cell   = take((size_t)2048 * 256);
  float* attn_cell= take((size_t)2048 * 256);
  float* heads_all= take((size_t)8 * 1024 * 256);
  float* tails_all= take((size_t)8 * 1024 * 256);
  float* qkv_drug = take((size_t)8192 * 768);   // reused heads -> tails
  float* o_drug   = take((size_t)8192 * 256);   // reused
  float* ao_drug  = take((size_t)8192 * 256);   // reused
  float* xc       = take((size_t)1024 * 768);
  float* h1       = take((size_t)1024 * 512);   // reused x3
  float* h2       = take((size_t)1024 * 256);   // reused x3
  float* xc1      = take((size_t)1024 * 768);
  float* xc2      = take((size_t)1024 * 768);
  float* subt     = take((size_t)1024);
  float* logits   = take((size_t)1024 * 1024);
  float* rloss    = take((size_t)1024);

  // --- cell learnable path: gather -> mean/max -> linear -> LN
  gather_agg_kernel<<<1024, 256, 0, stream>>>(protein_emb, neighbors, agg_in);
  launch_gemm(stream, agg_in, agg_w, agg_b, pre_cell, 1024, 256, 1024, 0, 0);
  layernorm256_kernel<<<1024, 256, 0, stream>>>(pre_cell, ln_g, ln_b, cell_all);

  // --- mashup path: L2 normalize -> linear -> same LN
  rownorm_kernel<<<1024, 256, 0, stream>>>(cell_mashup, mash_n, 800, 1e-12f);
  launch_gemm(stream, mash_n, mashup_w, mashup_b, pre_cell, 1024, 256, 800, 0, 0);
  layernorm256_kernel<<<1024, 256, 0, stream>>>(pre_cell, ln_g, ln_b,
                                                cell_all + (size_t)1024 * 256);

  // --- cell MHA over L=2, then mean -> xc[:,512:768]
  launch_gemm(stream, cell_all, ca_in_w, ca_in_b, qkv_cell, 2048, 768, 256, 1, 0);
  attn_small_kernel<2><<<1024, 256, 0, stream>>>(qkv_cell, o_cell, nullptr, 1024);
  launch_gemm(stream, o_cell, ca_out_w, ca_out_b, attn_cell, 2048, 256, 256, 1, 0);
  mean2_kernel<<<1024, 256, 0, stream>>>(attn_cell, xc, 768, 512, 1024);

  // --- drug modality embeddings -> heads_all / tails_all [8,1024,256]
  launch_gemm(stream, drug_a, t_w, t_b, heads_all, 1024, 256, 768, 0, 0);
  launch_gemm(stream, drug_b, t_w, t_b, tails_all, 1024, 256, 768, 0, 0);
  launch_gemm(stream, dv_drug_a, g_w, g_b, heads_all + (size_t)1024 * 256,
              1024, 256, 300, 0, 0);
  launch_gemm(stream, dv_drug_b, g_w, g_b, tails_all + (size_t)1024 * 256,
              1024, 256, 300, 0, 0);
  for (int m = 0; m < NOTH; ++m) {
    launch_gemm(stream, modal_a + (size_t)m * 1024 * 512,
                modal_w + (size_t)m * 512 * 256, modal_bias + (size_t)m * 256,
                heads_all + (size_t)(2 + m) * 1024 * 256, 1024, 256, 512, 0, 0);
    launch_gemm(stream, modal_b_in + (size_t)m * 1024 * 512,
                modal_w + (size_t)m * 512 * 256, modal_bias + (size_t)m * 256,
                tails_all + (size_t)(2 + m) * 1024 * 256, 1024, 256, 512, 0, 0);
  }

  // --- drug MHA (heads) + masked pool -> xc[:,0:256]
  launch_gemm(stream, heads_all, ma_in_w, ma_in_b, qkv_drug, 8192, 768, 256, 1, 0);
  attn_small_kernel<8><<<1024, 256, 0, stream>>>(qkv_drug, o_drug, heads_mask, 1024);
  launch_gemm(stream, o_drug, ma_out_w, ma_out_b, ao_drug, 8192, 256, 256, 1, 0);
  masked_pool_kernel<<<1024, 256, 0, stream>>>(ao_drug, heads_mask, xc, 768, 0, 1024);

  // --- drug MHA (tails) + masked pool -> xc[:,256:512]
  launch_gemm(stream, tails_all, ma_in_w, ma_in_b, qkv_drug, 8192, 768, 256, 1, 0);
  attn_small_kernel<8><<<1024, 256, 0, stream>>>(qkv_drug, o_drug, tail_mask, 1024);
  launch_gemm(stream, o_drug, ma_out_w, ma_out_b, ao_drug, 8192, 256, 256, 1, 0);
  masked_pool_kernel<<<1024, 256, 0, stream>>>(ao_drug, tail_mask, xc, 768, 256, 1024);

  // --- contrastive views (raw, pre-MHA embeddings + pre-MHA cell)
  build_view_kernel<<<1024, 256, 0, stream>>>(heads_all, tails_all, cell_all,
                                              ra1, rb1, cell_idx1, xc1, 1024);
  build_view_kernel<<<1024, 256, 0, stream>>>(heads_all, tails_all, cell_all,
                                              ra2, rb2, cell_idx2, xc2, 1024);

  // --- classifier: main output
  launch_gemm(stream, xc, c1_w, c1_b, h1, 1024, 512, 768, 0, 1);
  launch_gemm(stream, h1, c2a_w, c2a_b, h2, 1024, 256, 512, 0, 1);
  gemv_rows_kernel<<<128, 256, 0, stream>>>(h2, c2b_w, c2b_b, nullptr, 1.f,
                                            out, 256, 1024);
  // --- classifier: sub1
  launch_gemm(stream, xc1, c1_w, c1_b, h1, 1024, 512, 768, 0, 1);
  launch_gemm(stream, h1, c2a_w, c2a_b, h2, 1024, 256, 512, 0, 1);
  gemv_rows_kernel<<<128, 256, 0, stream>>>(h2, c2b_w, c2b_b, nullptr, 1.f,
                                            subt, 256, 1024);
  // --- classifier: sub2, fused 0.5*(sub1+sub2) -> out[1025:]
  launch_gemm(stream, xc2, c1_w, c1_b, h1, 1024, 512, 768, 0, 1);
  launch_gemm(stream, h1, c2a_w, c2a_b, h2, 1024, 256, 512, 0, 1);
  gemv_rows_kernel<<<128, 256, 0, stream>>>(h2, c2b_w, c2b_b, subt, 0.5f,
                                            out + 1025, 256, 1024);

  // --- cosine contrastive loss (raw views, normalized here; in-place OK now)
  rownorm_kernel<<<1024, 256, 0, stream>>>(xc1, xc1, 768, 1e-30f);
  rownorm_kernel<<<1024, 256, 0, stream>>>(xc2, xc2, 768, 1e-30f);
  launch_gemm(stream, xc1, xc2, nullptr, logits, 1024, 1024, 768, 1, 0);
  row_lse_loss_kernel<<<1024, 256, 0, stream>>>(logits, rloss, 1024);
  mean_reduce_kernel<<<1, 256, 0, stream>>>(rloss, out + 1024, 1024);
}